// GraphPartitionNet_61770219651287
// MI455X (gfx1250) — compile-verified
//
#include <hip/hip_runtime.h>
#include <hip/hip_bf16.h>

// ---------------------------------------------------------------------------
// GraphSAGE (4 layers) + MLP head for MI455X (gfx1250, wave32).
// Dense node transforms use V_WMMA_F32_16X16X4_F32 (full fp32 precision;
// FLOPs are roofline-irrelevant here).  Edge aggregation = coalesced gather
// + L2-resident global_atomic_add_f32 scatter.
// ---------------------------------------------------------------------------

constexpr int N_NODES = 100000;
constexpr int N_EDGES = 3200000;
constexpr int HID     = 16;
constexpr int DEC     = 32;
constexpr float BN_EPS = 1e-5f;

typedef float v2f __attribute__((ext_vector_type(2)));
typedef float v8f __attribute__((ext_vector_type(8)));

// ---------------------------------------------------------------- utilities
__global__ void zero_f32_kernel(float* __restrict__ p, int n) {
  int i = blockIdx.x * blockDim.x + threadIdx.x;
  if (i < n) p[i] = 0.0f;
}

// cnt[dst] += 1 over all edges (edge_index row-major [2,E]: src row 0, dst row 1)
__global__ void degree_kernel(const int* __restrict__ ei, float* __restrict__ cnt) {
  int e = blockIdx.x * blockDim.x + threadIdx.x;
  if (e < N_EDGES) {
    __builtin_prefetch(&ei[N_EDGES + e + 8192], 0, 0);
    atomicAdd(&cnt[ei[N_EDGES + e]], 1.0f);
  }
}

__global__ void invcnt_kernel(float* __restrict__ c) {
  int i = blockIdx.x * blockDim.x + threadIdx.x;
  if (i < N_NODES) c[i] = 1.0f / fmaxf(c[i], 1.0f);
}

// Layer-1 aggregation: scalar features, agg[dst] += x[src]
__global__ void agg_scalar_kernel(const int* __restrict__ ei, const float* __restrict__ x,
                                  float* __restrict__ agg) {
  int e = blockIdx.x * blockDim.x + threadIdx.x;
  if (e < N_EDGES) {
    __builtin_prefetch(&ei[e + 8192], 0, 0);
    atomicAdd(&agg[ei[N_EDGES + e]], x[ei[e]]);
  }
}

// Layer-1 transform fused with BatchNorm(eval) + ReLU.  1->16 width: VALU is right.
__global__ void layer1_bn_relu_kernel(const float* __restrict__ x, const float* __restrict__ agg,
                                      const float* __restrict__ inv,
                                      const float* __restrict__ W1l, const float* __restrict__ b1l,
                                      const float* __restrict__ W1r,
                                      const float* __restrict__ gamma, const float* __restrict__ beta,
                                      const float* __restrict__ mu, const float* __restrict__ var,
                                      float* __restrict__ out) {
  int i = blockIdx.x * blockDim.x + threadIdx.x;
  if (i >= N_NODES * HID) return;
  int n = i >> 4, f = i & 15;
  float a = agg[n] * inv[n];
  float v = a * W1l[f] + b1l[f] + x[n] * W1r[f];
  v = (v - mu[f]) * rsqrtf(var[f] + BN_EPS) * gamma[f] + beta[f];
  out[i] = fmaxf(v, 0.0f);
}

// 16-wide feature aggregation: agg[dst*16+f] += h[src*16+f].
// 16 consecutive lanes share one edge -> 64B contiguous gather, L2 atomics scatter.
__global__ void agg_feat16_kernel(const int* __restrict__ ei, const float* __restrict__ h,
                                  float* __restrict__ agg) {
  int t = blockIdx.x * blockDim.x + threadIdx.x;   // E*16 = 51.2M < 2^31
  int e = t >> 4, f = t & 15;
  if (e < N_EDGES) {
    __builtin_prefetch(&ei[e + 4096], 0, 0);
    int s = ei[e], d = ei[N_EDGES + e];
    atomicAdd(&agg[d * HID + f], h[s * HID + f]);
  }
}

// ---------------------------------------------------------------- WMMA GEMM
// One wave32 handles a 16-node tile.
//   SAGE=true : out = act( (agg * inv) @ Wl^T + bl + hin @ Wr^T )
//   SAGE=false: out = act( hin @ Wl^T + bl )                     (decoder)
// V_WMMA_F32_16X16X4_F32 fragment layouts (ISA 7.12.2):
//   A (16x4, MxK): lane l holds A[l&15][k0], A[l&15][k0+1], k0 = 2*(l>>4)
//   B (4x16, KxN): lane l holds B[k0][l&15], B[k0+1][l&15]   (mirrors A / SWMMAC B pattern)
//   C/D (16x16)  : VGPR j, lane l -> D[j + 8*(l>>4)][l&15]
template <int KIN, int NOUT, bool RELU, bool SAGE>
__global__ void node_gemm_wmma_kernel(const float* __restrict__ hin, const float* __restrict__ agg,
                                      const float* __restrict__ inv,
                                      const float* __restrict__ Wl, const float* __restrict__ bl,
                                      const float* __restrict__ Wr, float* __restrict__ out) {
  const int lane  = threadIdx.x & 31;
  const int wave  = threadIdx.x >> 5;
  const int tile  = blockIdx.x * (blockDim.x >> 5) + wave;
  if (tile * 16 >= N_NODES) return;              // wave-uniform: EXEC stays all-1 for WMMA

  const int m     = lane & 15;
  const int khalf = lane >> 4;
  const int arow  = tile * 16 + m;               // row this lane feeds into the A fragment
  float ivr = 0.0f;
  if constexpr (SAGE) ivr = inv[arow];

#pragma unroll
  for (int nb = 0; nb < NOUT; nb += 16) {
    const int ncol = nb + m;                     // this lane's output column (N index)
    v8f c;
    const float bias = bl[ncol];
#pragma unroll
    for (int j = 0; j < 8; ++j) c[j] = bias;     // bias broadcast into accumulator

#pragma unroll
    for (int kc = 0; kc < KIN; kc += 4) {
      const int k0 = kc + 2 * khalf;
      v2f a, b;
      if constexpr (SAGE) {
        // neighbor-mean path: (agg * inv) @ Wl^T
        a.x = agg[arow * KIN + k0] * ivr;
        a.y = agg[arow * KIN + k0 + 1] * ivr;
        b.x = Wl[ncol * KIN + k0];               // B[k][n] = Wl[n][k]
        b.y = Wl[ncol * KIN + k0 + 1];
        c = __builtin_amdgcn_wmma_f32_16x16x4_f32(false, a, false, b, (short)0, c, false, false);
        // self path: hin @ Wr^T
        a.x = hin[arow * KIN + k0];
        a.y = hin[arow * KIN + k0 + 1];
        b.x = Wr[ncol * KIN + k0];
        b.y = Wr[ncol * KIN + k0 + 1];
        c = __builtin_amdgcn_wmma_f32_16x16x4_f32(false, a, false, b, (short)0, c, false, false);
      } else {
        a.x = hin[arow * KIN + k0];
        a.y = hin[arow * KIN + k0 + 1];
        b.x = Wl[ncol * KIN + k0];
        b.y = Wl[ncol * KIN + k0 + 1];
        c = __builtin_amdgcn_wmma_f32_16x16x4_f32(false, a, false, b, (short)0, c, false, false);
      }
    }

#pragma unroll
    for (int j = 0; j < 8; ++j) {
      const int orow = tile * 16 + j + 8 * khalf;
      float v = c[j];
      if constexpr (RELU) v = fmaxf(v, 0.0f);
      out[orow * NOUT + ncol] = v;
    }
  }
}

// Final 32->2 projection + softmax (tiny; VALU kernel).
__global__ void head_softmax_kernel(const float* __restrict__ hd, const float* __restrict__ Wd2,
                                    const float* __restrict__ bd2, float* __restrict__ out) {
  int n = blockIdx.x * blockDim.x + threadIdx.x;
  if (n >= N_NODES) return;
  float z0 = bd2[0], z1 = bd2[1];
#pragma unroll
  for (int k = 0; k < DEC; ++k) {
    float h = hd[n * DEC + k];
    z0 += h * Wd2[k];
    z1 += h * Wd2[DEC + k];
  }
  float mx = fmaxf(z0, z1);
  float e0 = __expf(z0 - mx), e1 = __expf(z1 - mx);
  float s = 1.0f / (e0 + e1);
  out[n * 2 + 0] = e0 * s;
  out[n * 2 + 1] = e1 * s;
}

// ---------------------------------------------------------------- launcher
static inline int cdiv(long long a, int b) { return (int)((a + b - 1) / b); }

extern "C" void kernel_launch(void* const* d_in, const int* in_sizes, int n_in,
                              void* d_out, int out_size, void* d_ws, size_t ws_size,
                              hipStream_t stream) {
  const float* x    = (const float*)d_in[0];
  const int*   ei   = (const int*)d_in[1];      // [2,E] row-major: src row, dst row
  const float* W1l  = (const float*)d_in[2];
  const float* b1l  = (const float*)d_in[3];
  const float* W1r  = (const float*)d_in[4];
  const float* bng  = (const float*)d_in[5];
  const float* bnb  = (const float*)d_in[6];
  const float* bnm  = (const float*)d_in[7];
  const float* bnv  = (const float*)d_in[8];
  const float* W2l  = (const float*)d_in[9];
  const float* b2l  = (const float*)d_in[10];
  const float* W2r  = (const float*)d_in[11];
  const float* W3l  = (const float*)d_in[12];
  const float* b3l  = (const float*)d_in[13];
  const float* W3r  = (const float*)d_in[14];
  const float* W4l  = (const float*)d_in[15];
  const float* b4l  = (const float*)d_in[16];
  const float* W4r  = (const float*)d_in[17];
  const float* Wd1  = (const float*)d_in[18];
  const float* bd1  = (const float*)d_in[19];
  const float* Wd2  = (const float*)d_in[20];
  const float* bd2  = (const float*)d_in[21];
  float* outp = (float*)d_out;

  // scratch layout (floats): inv | hA[N,16] | hB[N,16] | agg[N,16] | h3[N,32] | hd[N,32]
  float* ws  = (float*)d_ws;
  float* inv = ws;
  float* hA  = inv + N_NODES;
  float* hB  = hA + (size_t)N_NODES * HID;
  float* agg = hB + (size_t)N_NODES * HID;
  float* h3  = agg + (size_t)N_NODES * HID;
  float* hd  = h3 + (size_t)N_NODES * DEC;

  const int ZB = 256, EB = 256;
  const int tiles  = N_NODES / 16;              // 100000 / 16 = 6250 exactly
  const int wblk   = 128;                        // 4 waves per block
  const int wgrid  = cdiv(tiles, wblk / 32);

  // degree -> inv_cnt
  zero_f32_kernel<<<cdiv(N_NODES, ZB), ZB, 0, stream>>>(inv, N_NODES);
  degree_kernel<<<cdiv(N_EDGES, EB), EB, 0, stream>>>(ei, inv);
  invcnt_kernel<<<cdiv(N_NODES, ZB), ZB, 0, stream>>>(inv);

  // layer 1 (scalar agg) + BN + ReLU -> hA
  zero_f32_kernel<<<cdiv(N_NODES, ZB), ZB, 0, stream>>>(agg, N_NODES);
  agg_scalar_kernel<<<cdiv(N_EDGES, EB), EB, 0, stream>>>(ei, x, agg);
  layer1_bn_relu_kernel<<<cdiv((long long)N_NODES * HID, ZB), ZB, 0, stream>>>(
      x, agg, inv, W1l, b1l, W1r, bng, bnb, bnm, bnv, hA);

  // layer 2: hA -> hB
  zero_f32_kernel<<<cdiv((long long)N_NODES * HID, ZB), ZB, 0, stream>>>(agg, N_NODES * HID);
  agg_feat16_kernel<<<cdiv((long long)N_EDGES * HID, EB), EB, 0, stream>>>(ei, hA, agg);
  node_gemm_wmma_kernel<16, 16, true, true><<<wgrid, wblk, 0, stream>>>(hA, agg, inv, W2l, b2l, W2r, hB);

  // layer 3: hB -> hA
  zero_f32_kernel<<<cdiv((long long)N_NODES * HID, ZB), ZB, 0, stream>>>(agg, N_NODES * HID);
  agg_feat16_kernel<<<cdiv((long long)N_EDGES * HID, EB), EB, 0, stream>>>(ei, hB, agg);
  node_gemm_wmma_kernel<16, 16, true, true><<<wgrid, wblk, 0, stream>>>(hB, agg, inv, W3l, b3l, W3r, hA);

  // layer 4 (16 -> 32, no ReLU): hA -> h3
  zero_f32_kernel<<<cdiv((long long)N_NODES * HID, ZB), ZB, 0, stream>>>(agg, N_NODES * HID);
  agg_feat16_kernel<<<cdiv((long long)N_EDGES * HID, EB), EB, 0, stream>>>(ei, hA, agg);
  node_gemm_wmma_kernel<16, 32, false, true><<<wgrid, wblk, 0, stream>>>(hA, agg, inv, W4l, b4l, W4r, h3);

  // decoder dense 32->32 + ReLU: h3 -> hd
  node_gemm_wmma_kernel<32, 32, true, false><<<wgrid, wblk, 0, stream>>>(h3, nullptr, nullptr, Wd1, bd1, nullptr, hd);

  // head: 32 -> 2 + softmax
  head_softmax_kernel<<<cdiv(N_NODES, ZB), ZB, 0, stream>>>(hd, Wd2, bd2, outp);
}